// GraphMultiHeadAttention_10934986735977
// MI455X (gfx1250) — compile-verified
//
#include <hip/hip_runtime.h>

// Problem constants (match reference)
#define Bq  8
#define Nq  1024
#define Dq  256
#define Hq  4
#define HOq 64
#define Tq  5
#define NEGV (-1e9f)
#define EPSV (1e-6f)

typedef float v2f __attribute__((ext_vector_type(2)));
typedef float v8f __attribute__((ext_vector_type(8)));

__device__ __forceinline__ float sel5(int t, float e0, float e1, float e2, float e3, float e4) {
    return (t == 0) ? e0 : (t == 1) ? e1 : (t == 2) ? e2 : (t == 3) ? e3 : e4;
}

// ---------------------------------------------------------------------------
// Kernel 1: h[b,h,n,e] = X[b,n,:] @ W[h,:,e]  via V_WMMA_F32_16X16X4_F32
// one wave (32 lanes) per 16x16 output tile, K-loop over D in steps of 4
// ---------------------------------------------------------------------------
__global__ void __launch_bounds__(32) k_proj(const float* __restrict__ X,
                                             const float* __restrict__ Wm,
                                             float* __restrict__ hws) {
    const int lane = threadIdx.x;
    const int half = lane >> 4;      // 0: lanes 0-15, 1: lanes 16-31
    const int lm   = lane & 15;
    const int i0   = blockIdx.x * 16;
    const int ho0  = blockIdx.y * 16;
    const int bh   = blockIdx.z;
    const int b    = bh / Hq, hh = bh % Hq;

    const float* xr = X + ((size_t)b * Nq + (size_t)(i0 + lm)) * Dq;   // A row per lane
    const float* wp = Wm + (size_t)hh * Dq * HOq + ho0 + lm;           // B col per lane

    v8f c = {};
#pragma unroll 4
    for (int k = 0; k < Dq; k += 4) {
        const int kk = k + half * 2;
        // A-frag (16x4 f32): lane holds A[lm, kk], A[lm, kk+1] -> 8B load
        float2 axy = *(const float2*)(xr + kk);
        v2f a; a.x = axy.x; a.y = axy.y;
        // B-frag (4x16 f32): lane holds B[kk, ho0+lm], B[kk+1, ho0+lm]
        v2f bb;
        bb.x = wp[(size_t)kk * HOq];
        bb.y = wp[(size_t)(kk + 1) * HOq];
        c = __builtin_amdgcn_wmma_f32_16x16x4_f32(false, a, false, bb,
                                                  (short)0, c, false, false);
    }
    // C/D layout: VGPR r holds row (r + 8*half), col lm
    float* hp = hws + ((size_t)bh * Nq + i0) * HOq + ho0 + lm;
#pragma unroll
    for (int r = 0; r < 8; ++r)
        hp[(size_t)(r + 8 * half) * HOq] = c[r];
}

// ---------------------------------------------------------------------------
// Kernel 2: e_src[b,h,t,n] = h[b,h,n,:].a_src[h,t,:] ; same for e_dst
// ---------------------------------------------------------------------------
__global__ void __launch_bounds__(256) k_escore(const float* __restrict__ hws,
                                                const float* __restrict__ asrc,
                                                const float* __restrict__ adst,
                                                float* __restrict__ es,
                                                float* __restrict__ ed) {
    int idx = blockIdx.x * 256 + threadIdx.x;          // over B*H*T*N
    if (idx >= Bq * Hq * Tq * Nq) return;
    const int n  = idx % Nq;
    const int t  = (idx / Nq) % Tq;
    const int bh = idx / (Nq * Tq);
    const int hh = bh % Hq;
    const float4* hv = (const float4*)(hws + ((size_t)bh * Nq + n) * HOq);
    const float4* sv = (const float4*)(asrc + ((size_t)hh * Tq + t) * HOq);
    const float4* dv = (const float4*)(adst + ((size_t)hh * Tq + t) * HOq);
    float ss = 0.f, sd = 0.f;
#pragma unroll
    for (int q = 0; q < HOq / 4; ++q) {
        float4 hx = hv[q], sx = sv[q], dx = dv[q];
        ss += hx.x * sx.x + hx.y * sx.y + hx.z * sx.z + hx.w * sx.w;
        sd += hx.x * dx.x + hx.y * dx.y + hx.z * dx.z + hx.w * dx.w;
    }
    es[idx] = ss;
    ed[idx] = sd;
}

// ---------------------------------------------------------------------------
// Kernel 3: softmax pass A -- per-row running max & sum(exp) (online, wave32)
// ---------------------------------------------------------------------------
__global__ void __launch_bounds__(32) k_softstats(const int* __restrict__ adj,
                                                  const float* __restrict__ es,
                                                  const float* __restrict__ ed,
                                                  float* __restrict__ rmax,
                                                  float* __restrict__ rsum) {
    const int lane  = threadIdx.x;
    const int rowid = blockIdx.x;          // over B*H*N
    const int i  = rowid % Nq;
    const int bh = rowid / Nq;
    const int b  = bh / Hq;

    const float e0 = es[((size_t)bh * Tq + 0) * Nq + i];
    const float e1 = es[((size_t)bh * Tq + 1) * Nq + i];
    const float e2 = es[((size_t)bh * Tq + 2) * Nq + i];
    const float e3 = es[((size_t)bh * Tq + 3) * Nq + i];
    const float e4 = es[((size_t)bh * Tq + 4) * Nq + i];
    const int*   ar  = adj + ((size_t)b * Nq + i) * Nq;
    const float* edb = ed + (size_t)bh * Tq * Nq;

    float m = -INFINITY, s = 0.f;
    for (int j = lane; j < Nq; j += 32) {
        int a = ar[j];
        int t = (a > 0) ? (a - 1) : 0;                 // branchless, safe index
        float sc = sel5(t, e0, e1, e2, e3, e4) + edb[(size_t)t * Nq + j];
        float w = (a == 0) ? NEGV : (sc > 0.f ? sc : 0.01f * sc);   // leaky+mask
        float mn = fmaxf(m, w);
        s = s * __expf(m - mn) + __expf(w - mn);
        m = mn;
    }
    // wave32 reduction (max with rescaled sums)
#pragma unroll
    for (int off = 16; off >= 1; off >>= 1) {
        float mo = __shfl_xor(m, off, 32);
        float so = __shfl_xor(s, off, 32);
        float mn = fmaxf(m, mo);
        s = s * __expf(m - mn) + so * __expf(mo - mn);
        m = mn;
    }
    if (lane == 0) { rmax[rowid] = m; rsum[rowid] = s; }
}

// ---------------------------------------------------------------------------
// Kernel 4: fused  out = LN(relu(P@h + bias) + h)  with P recomputed on the
// fly per 16x4 A-frag, accumulated with 4x V_WMMA_F32_16X16X4_F32 (HO=64)
// ---------------------------------------------------------------------------
__global__ void __launch_bounds__(32) k_attn(const int* __restrict__ adj,
                                             const float* __restrict__ hws,
                                             const float* __restrict__ es,
                                             const float* __restrict__ ed,
                                             const float* __restrict__ rmax,
                                             const float* __restrict__ rsum,
                                             const float* __restrict__ bias,
                                             const float* __restrict__ gamma,
                                             const float* __restrict__ beta,
                                             float* __restrict__ out) {
    __shared__ float tile[16 * 68];        // 16 rows x HO, padded (stride 68 -> no bank conflicts)
    const int lane = threadIdx.x;
    const int half = lane >> 4;
    const int lm   = lane & 15;
    const int i0   = blockIdx.x * 16;
    const int bh   = blockIdx.y;
    const int b    = bh / Hq, hh = bh % Hq;
    const int row  = i0 + lm;              // A-frag row owned by this lane

    const float e0 = es[((size_t)bh * Tq + 0) * Nq + row];
    const float e1 = es[((size_t)bh * Tq + 1) * Nq + row];
    const float e2 = es[((size_t)bh * Tq + 2) * Nq + row];
    const float e3 = es[((size_t)bh * Tq + 3) * Nq + row];
    const float e4 = es[((size_t)bh * Tq + 4) * Nq + row];
    const float rm   = rmax[(size_t)bh * Nq + row];
    const float rinv = 1.f / rsum[(size_t)bh * Nq + row];

    const int*   ar  = adj + ((size_t)b * Nq + row) * Nq;
    const float* edb = ed + (size_t)bh * Tq * Nq;
    const float* hb  = hws + (size_t)bh * Nq * HOq;

    v8f c0 = {}, c1 = {}, c2 = {}, c3 = {};
    for (int j0 = 0; j0 < Nq; j0 += 4) {
        const int jj = j0 + half * 2;
        int2 av = *(const int2*)(ar + jj);             // adj[row, jj..jj+1], 8B aligned

        float p0, p1;
        {
            int a = av.x; int t = (a > 0) ? (a - 1) : 0;
            float sc = sel5(t, e0, e1, e2, e3, e4) + edb[(size_t)t * Nq + jj];
            float w = (a == 0) ? NEGV : (sc > 0.f ? sc : 0.01f * sc);
            p0 = __expf(w - rm) * rinv;
        }
        {
            int a = av.y; int t = (a > 0) ? (a - 1) : 0;
            float sc = sel5(t, e0, e1, e2, e3, e4) + edb[(size_t)t * Nq + jj + 1];
            float w = (a == 0) ? NEGV : (sc > 0.f ? sc : 0.01f * sc);
            p1 = __expf(w - rm) * rinv;
        }
        v2f ap; ap.x = p0; ap.y = p1;

        const float* hr0 = hb + (size_t)jj * HOq + lm;     // B rows jj, jj+1
        const float* hr1 = hr0 + HOq;
        v2f b0, b1, b2, b3;
        b0.x = hr0[0];  b0.y = hr1[0];
        b1.x = hr0[16]; b1.y = hr1[16];
        b2.x = hr0[32]; b2.y = hr1[32];
        b3.x = hr0[48]; b3.y = hr1[48];

        c0 = __builtin_amdgcn_wmma_f32_16x16x4_f32(false, ap, false, b0, (short)0, c0, false, false);
        c1 = __builtin_amdgcn_wmma_f32_16x16x4_f32(false, ap, false, b1, (short)0, c1, false, false);
        c2 = __builtin_amdgcn_wmma_f32_16x16x4_f32(false, ap, false, b2, (short)0, c2, false, false);
        c3 = __builtin_amdgcn_wmma_f32_16x16x4_f32(false, ap, false, b3, (short)0, c3, false, false);
    }

    // spill accumulators (row r+8*half, col lm within each 16-wide slice) to LDS
#pragma unroll
    for (int r = 0; r < 8; ++r) {
        const int m = r + 8 * half;
        tile[m * 68 +  0 + lm] = c0[r];
        tile[m * 68 + 16 + lm] = c1[r];
        tile[m * 68 + 32 + lm] = c2[r];
        tile[m * 68 + 48 + lm] = c3[r];
    }
    __syncthreads();

    // epilogue: bias -> relu -> +h residual -> per-head LayerNorm -> concat store
    if (lane < 16) {
        const int r = lane;
        const float* hres = hb + (size_t)(i0 + r) * HOq;
        float sum = 0.f, sumsq = 0.f;
#pragma unroll 8
        for (int e = 0; e < HOq; ++e) {
            float v = tile[r * 68 + e] + bias[hh * HOq + e];
            v = fmaxf(v, 0.f);
            v += hres[e];
            tile[r * 68 + e] = v;
            sum += v; sumsq += v * v;
        }
        const float mu   = sum * (1.f / HOq);
        const float var  = sumsq * (1.f / HOq) - mu * mu;   // biased variance (torch)
        const float rstd = rsqrtf(var + EPSV);
        float* op = out + ((size_t)b * Nq + (i0 + r)) * (Hq * HOq) + hh * HOq;
#pragma unroll 8
        for (int e = 0; e < HOq; ++e)
            op[e] = (tile[r * 68 + e] - mu) * rstd * gamma[hh * HOq + e] + beta[hh * HOq + e];
    }
}

// ---------------------------------------------------------------------------
extern "C" void kernel_launch(void* const* d_in, const int* in_sizes, int n_in,
                              void* d_out, int out_size, void* d_ws, size_t ws_size,
                              hipStream_t stream) {
    const float* nodes = (const float*)d_in[0];   // [B,N,D]
    const int*   adj   = (const int*)d_in[1];     // [B,N,N]
    const float* Wm    = (const float*)d_in[2];   // [H,D,HO]
    const float* asrc  = (const float*)d_in[3];   // [H,T,HO]
    const float* adst  = (const float*)d_in[4];   // [H,T,HO]
    const float* bias  = (const float*)d_in[5];   // [H,HO]
    const float* gamma = (const float*)d_in[6];   // [H,HO]
    const float* beta  = (const float*)d_in[7];   // [H,HO]
    float* out = (float*)d_out;                   // [B,N,H*HO]

    // workspace layout (f32): ~10 MB total
    float* h_ws = (float*)d_ws;                                   // B*H*N*HO
    float* esrc = h_ws + (size_t)Bq * Hq * Nq * HOq;              // B*H*T*N
    float* edst = esrc + (size_t)Bq * Hq * Tq * Nq;               // B*H*T*N
    float* rmx  = edst + (size_t)Bq * Hq * Tq * Nq;               // B*H*N
    float* rsm  = rmx  + (size_t)Bq * Hq * Nq;                    // B*H*N

    k_proj<<<dim3(Nq / 16, HOq / 16, Bq * Hq), 32, 0, stream>>>(nodes, Wm, h_ws);
    k_escore<<<(Bq * Hq * Tq * Nq + 255) / 256, 256, 0, stream>>>(h_ws, asrc, adst, esrc, edst);
    k_softstats<<<Bq * Hq * Nq, 32, 0, stream>>>(adj, esrc, edst, rmx, rsm);
    k_attn<<<dim3(Nq / 16, Bq * Hq), 32, 0, stream>>>(adj, h_ws, esrc, edst, rmx, rsm,
                                                      bias, gamma, beta, out);
}